// SignedGCN_75204877353504
// MI455X (gfx1250) — compile-verified
//
#include <hip/hip_runtime.h>

typedef __attribute__((ext_vector_type(2))) float v2f;
typedef __attribute__((ext_vector_type(8))) float v8f;

// ---------------------------------------------------------------------------
// FP32 WMMA 16x16x4: D = A(16x4) x B(4x16) + C, full fp32 precision.
// ---------------------------------------------------------------------------
__device__ __forceinline__ v8f wmma_f32(v2f a, v2f b, v8f c) {
  return __builtin_amdgcn_wmma_f32_16x16x4_f32(
      /*neg_a=*/false, a, /*neg_b=*/false, b,
      /*c_mod=*/(short)0, c, /*reuse_a=*/false, /*reuse_b=*/false);
}

// A fragment: 16x4 tile of a row-major matrix, leading dim 64.
// Lanes 0-15: K=k0,k0+1 ; lanes 16-31: K=k0+2,k0+3. One b64 load (8B aligned).
__device__ __forceinline__ v2f load_a64(const float* M, int row, int k0, int lane) {
  int kk = k0 + ((lane >> 4) << 1);
  return *(const v2f*)(M + (size_t)row * 64 + kk);
}

// B fragment from K-paired packed weights (32 cols):
//   packed[((k>>1)*32 + col)*2 + (k&1)] = W[k][col]
// so {W[kk][col], W[kk+1][col]} is one contiguous b64 (ds_load_b64 from LDS).
__device__ __forceinline__ v2f load_b_pk(const float* Wp, int k0, int n0, int lane) {
  int pair = (k0 >> 1) + (lane >> 4);
  return *(const v2f*)(Wp + ((size_t)(pair * 32 + n0 + (lane & 15)) << 1));
}

// ---------------------------------------------------------------------------
// Utility kernels
// ---------------------------------------------------------------------------
__global__ void zero_kernel(float* __restrict__ p, long long count) {
  long long i = (long long)blockIdx.x * blockDim.x + threadIdx.x;
  long long stride = (long long)gridDim.x * blockDim.x;
  for (; i < count; i += stride) p[i] = 0.0f;
}

__global__ void recip_kernel(float* __restrict__ a, float* __restrict__ b, int n) {
  int i = blockIdx.x * blockDim.x + threadIdx.x;
  if (i < n) {
    a[i] = 1.0f / fmaxf(a[i], 1.0f);
    b[i] = 1.0f / fmaxf(b[i], 1.0f);
  }
}

// Repack a K x 32 row-major weight matrix into the paired-K fragment layout.
__global__ void pack_weights_kernel(const float* __restrict__ W,
                                    float* __restrict__ Wp, int K) {
  int i = blockIdx.x * blockDim.x + threadIdx.x;  // over K*32 elements
  if (i >= K * 32) return;
  int k = i >> 5, col = i & 31;
  Wp[(((k >> 1) * 32 + col) << 1) + (k & 1)] = W[i];
}

// ---------------------------------------------------------------------------
// Layer-1 scatter: agg[dst][0:64] += x[src][0:64], fused degree count.
// 16 threads per edge, 4 features each (b128 gather -> 4 L2-resident atomics).
// ---------------------------------------------------------------------------
__global__ void __launch_bounds__(256) scatter1_kernel(
    const float* __restrict__ xf, const int* __restrict__ srcI,
    const int* __restrict__ dstI, float* __restrict__ agg,
    float* __restrict__ cnt, long long nE) {
  long long t = (long long)blockIdx.x * blockDim.x + threadIdx.x;
  long long e = t >> 4;
  if (e >= nE) return;
  int f4 = (int)(t & 15) << 2;
  if (f4 == 0) {  // one lane per edge: prefetch the index stream ahead
    __builtin_prefetch(srcI + e + 4096, 0, 1);
    __builtin_prefetch(dstI + e + 4096, 0, 1);
  }
  int s = srcI[e], d = dstI[e];
  const float4 v = *(const float4*)(xf + (size_t)s * 64 + f4);
  float* o = agg + (size_t)d * 64 + f4;
  atomicAdd(o + 0, v.x); atomicAdd(o + 1, v.y);
  atomicAdd(o + 2, v.z); atomicAdd(o + 3, v.w);
  if (f4 == 0) atomicAdd(cnt + d, 1.0f);  // fused degree count
}

// ---------------------------------------------------------------------------
// Layer-2 scatter over z halves, routed into aggA=[p1|p2], aggB=[n1|n2]:
//   pos edges: f<32 (zp) -> aggA[f]      ; f>=32 (zn) -> aggB[f-32]
//   neg edges: f<32 (zp) -> aggB[32+f]   ; f>=32 (zn) -> aggA[f]
// ---------------------------------------------------------------------------
__global__ void __launch_bounds__(256) scatter2_kernel(
    const float* __restrict__ zf, const int* __restrict__ srcI,
    const int* __restrict__ dstI, float* __restrict__ aggA,
    float* __restrict__ aggB, int isNeg, long long nE) {
  long long t = (long long)blockIdx.x * blockDim.x + threadIdx.x;
  long long e = t >> 4;
  if (e >= nE) return;
  int f4 = (int)(t & 15) << 2;
  if (f4 == 0) {
    __builtin_prefetch(srcI + e + 4096, 0, 1);
    __builtin_prefetch(dstI + e + 4096, 0, 1);
  }
  int s = srcI[e], d = dstI[e];
  const float4 v = *(const float4*)(zf + (size_t)s * 64 + f4);
  float* o;
  if (!isNeg) o = (f4 < 32) ? (aggA + (size_t)d * 64 + f4)
                            : (aggB + (size_t)d * 64 + (f4 - 32));
  else        o = (f4 < 32) ? (aggB + (size_t)d * 64 + (32 + f4))
                            : (aggA + (size_t)d * 64 + f4);
  atomicAdd(o + 0, v.x); atomicAdd(o + 1, v.y);
  atomicAdd(o + 2, v.z); atomicAdd(o + 3, v.w);
}

// ---------------------------------------------------------------------------
// Dense layer 1 (one wave32 per 16-node tile, fp32 WMMA):
//   z[:, 0:32] = relu(aggA*invp @ w_pl + x @ w_pr + b_p)
//   z[:,32:64] = relu(aggB*invn @ w_nl + x @ w_nr + b_n)
// Packed weights staged in LDS once per block (8 waves share).
// ---------------------------------------------------------------------------
__global__ void __launch_bounds__(256) dense1_kernel(
    const float* __restrict__ x, const float* __restrict__ aggA,
    const float* __restrict__ aggB, const float* __restrict__ invp,
    const float* __restrict__ invn, const float* __restrict__ pw,
    const float* __restrict__ b_p, const float* __restrict__ b_n,
    float* __restrict__ z, int n) {
  __shared__ float lw[8192];  // 4 packed 64x32 matrices, 32 KB
  for (int i = threadIdx.x; i < 2048; i += 256)
    reinterpret_cast<float4*>(lw)[i] = reinterpret_cast<const float4*>(pw)[i];
  __syncthreads();

  int wave = (int)((blockIdx.x * blockDim.x + threadIdx.x) >> 5);
  int lane = threadIdx.x & 31;
  int row0 = wave * 16;
  if (row0 >= n) return;

  const float* l_pl = lw;
  const float* l_pr = lw + 2048;
  const float* l_nl = lw + 4096;
  const float* l_nr = lw + 6144;

  int arow = row0 + (lane & 15); if (arow > n - 1) arow = n - 1;
  float sp = invp[arow], sn = invn[arow];
  v8f p0 = {0,0,0,0,0,0,0,0}, p1 = p0, q0 = p0, q1 = p0;
  for (int k0 = 0; k0 < 64; k0 += 4) {
    v2f ax = load_a64(x,    arow, k0, lane);
    v2f ap = load_a64(aggA, arow, k0, lane); ap.x *= sp; ap.y *= sp;
    v2f an = load_a64(aggB, arow, k0, lane); an.x *= sn; an.y *= sn;
    p0 = wmma_f32(ap, load_b_pk(l_pl, k0,  0, lane), p0);
    p0 = wmma_f32(ax, load_b_pk(l_pr, k0,  0, lane), p0);
    p1 = wmma_f32(ap, load_b_pk(l_pl, k0, 16, lane), p1);
    p1 = wmma_f32(ax, load_b_pk(l_pr, k0, 16, lane), p1);
    q0 = wmma_f32(an, load_b_pk(l_nl, k0,  0, lane), q0);
    q0 = wmma_f32(ax, load_b_pk(l_nr, k0,  0, lane), q0);
    q1 = wmma_f32(an, load_b_pk(l_nl, k0, 16, lane), q1);
    q1 = wmma_f32(ax, load_b_pk(l_nr, k0, 16, lane), q1);
  }
  int col = lane & 15;
  int rbase = (lane >> 4) << 3;
  float bp0 = b_p[col], bp1 = b_p[col + 16];
  float bn0 = b_n[col], bn1 = b_n[col + 16];
  for (int r = 0; r < 8; ++r) {
    int row = row0 + rbase + r;
    if (row < n) {
      float* o = z + (size_t)row * 64;
      o[col]      = fmaxf(p0[r] + bp0, 0.0f);
      o[col + 16] = fmaxf(p1[r] + bp1, 0.0f);
      o[col + 32] = fmaxf(q0[r] + bn0, 0.0f);
      o[col + 48] = fmaxf(q1[r] + bn1, 0.0f);
    }
  }
}

// ---------------------------------------------------------------------------
// Dense layer 2 (in-place on d_out; each wave owns its 16-row tile):
//   out_pos = [p1|p2]*inv @ w2_pl + zp @ w2_pr + b2_p   (cols 0:32)
//   out_neg = [n1|n2]*inv @ w2_nl + zn @ w2_nr + b2_n   (cols 32:64)
// Column halves of agg buffers scale by invp (k<32) / invn (k>=32).
// ---------------------------------------------------------------------------
__global__ void __launch_bounds__(256) dense2_kernel(
    float* zio, const float* __restrict__ aggA, const float* __restrict__ aggB,
    const float* __restrict__ invp, const float* __restrict__ invn,
    const float* __restrict__ pw, const float* __restrict__ b_p,
    const float* __restrict__ b_n, int n) {
  __shared__ float lw[6144];  // w2_pl(2048) w2_nl(2048) w2_pr(1024) w2_nr(1024)
  for (int i = threadIdx.x; i < 1536; i += 256)
    reinterpret_cast<float4*>(lw)[i] = reinterpret_cast<const float4*>(pw)[i];
  __syncthreads();

  int wave = (int)((blockIdx.x * blockDim.x + threadIdx.x) >> 5);
  int lane = threadIdx.x & 31;
  int row0 = wave * 16;
  if (row0 >= n) return;

  const float* l_pl = lw;
  const float* l_nl = lw + 2048;
  const float* l_pr = lw + 4096;
  const float* l_nr = lw + 5120;

  int arow = row0 + (lane & 15); if (arow > n - 1) arow = n - 1;
  float sp = invp[arow], sn = invn[arow];
  v8f p0 = {0,0,0,0,0,0,0,0}, p1 = p0, q0 = p0, q1 = p0;
  for (int k0 = 0; k0 < 64; k0 += 4) {
    float sA = (k0 < 32) ? sp : sn;
    v2f aA = load_a64(aggA, arow, k0, lane); aA.x *= sA; aA.y *= sA;
    v2f aB = load_a64(aggB, arow, k0, lane); aB.x *= sA; aB.y *= sA;
    p0 = wmma_f32(aA, load_b_pk(l_pl, k0,  0, lane), p0);
    p1 = wmma_f32(aA, load_b_pk(l_pl, k0, 16, lane), p1);
    q0 = wmma_f32(aB, load_b_pk(l_nl, k0,  0, lane), q0);
    q1 = wmma_f32(aB, load_b_pk(l_nl, k0, 16, lane), q1);
    if (k0 < 32) {
      v2f azp = load_a64(zio, arow, k0,      lane);
      v2f azn = load_a64(zio, arow, 32 + k0, lane);
      p0 = wmma_f32(azp, load_b_pk(l_pr, k0,  0, lane), p0);
      p1 = wmma_f32(azp, load_b_pk(l_pr, k0, 16, lane), p1);
      q0 = wmma_f32(azn, load_b_pk(l_nr, k0,  0, lane), q0);
      q1 = wmma_f32(azn, load_b_pk(l_nr, k0, 16, lane), q1);
    }
  }
  int col = lane & 15;
  int rbase = (lane >> 4) << 3;
  float bp0 = b_p[col], bp1 = b_p[col + 16];
  float bn0 = b_n[col], bn1 = b_n[col + 16];
  for (int r = 0; r < 8; ++r) {
    int row = row0 + rbase + r;
    if (row < n) {
      float* o = zio + (size_t)row * 64;
      o[col]      = fmaxf(p0[r] + bp0, 0.0f);
      o[col + 16] = fmaxf(p1[r] + bp1, 0.0f);
      o[col + 32] = fmaxf(q0[r] + bn0, 0.0f);
      o[col + 48] = fmaxf(q1[r] + bn1, 0.0f);
    }
  }
}

// ---------------------------------------------------------------------------
extern "C" void kernel_launch(void* const* d_in, const int* in_sizes, int n_in,
                              void* d_out, int out_size, void* d_ws, size_t ws_size,
                              hipStream_t stream) {
  const float* x     = (const float*)d_in[0];
  const int*   posE  = (const int*)d_in[1];
  const int*   negE  = (const int*)d_in[2];
  const float* w1_pl = (const float*)d_in[3];
  const float* w1_pr = (const float*)d_in[4];
  const float* b1_p  = (const float*)d_in[5];
  const float* w1_nl = (const float*)d_in[6];
  const float* w1_nr = (const float*)d_in[7];
  const float* b1_n  = (const float*)d_in[8];
  const float* w2_pl = (const float*)d_in[9];
  const float* w2_pr = (const float*)d_in[10];
  const float* b2_p  = (const float*)d_in[11];
  const float* w2_nl = (const float*)d_in[12];
  const float* w2_nr = (const float*)d_in[13];
  const float* b2_n  = (const float*)d_in[14];

  const int n = in_sizes[0] / 64;
  const long long nE = in_sizes[1] / 2;

  float* ws   = (float*)d_ws;
  float* invp = ws;                       // N   (counts, then 1/deg)
  float* invn = ws + n;                   // N
  float* aggA = ws + 2 * (size_t)n;       // N*64
  float* aggB = aggA + (size_t)n * 64;    // N*64
  size_t pwoff = (((size_t)n * 130) + 3) & ~(size_t)3;  // 16B-aligned
  float* pw1  = ws + pwoff;               // 4 x 2048 packed layer-1 weights
  float* pw2  = pw1 + 8192;               // 2048+2048+1024+1024 layer-2
  float* z    = (float*)d_out;            // N*64, also final output

  // 1) zero counts + agg buffers (contiguous region of 130*N floats)
  {
    long long cnt = (long long)n * 130;
    int blocks = (int)((cnt + 255) / 256); if (blocks > 32768) blocks = 32768;
    zero_kernel<<<blocks, 256, 0, stream>>>(ws, cnt);
  }
  // 1b) pack all weight matrices into paired-K fragment layout
  pack_weights_kernel<<<8, 256, 0, stream>>>(w1_pl, pw1,        64);
  pack_weights_kernel<<<8, 256, 0, stream>>>(w1_pr, pw1 + 2048, 64);
  pack_weights_kernel<<<8, 256, 0, stream>>>(w1_nl, pw1 + 4096, 64);
  pack_weights_kernel<<<8, 256, 0, stream>>>(w1_nr, pw1 + 6144, 64);
  pack_weights_kernel<<<8, 256, 0, stream>>>(w2_pl, pw2,        64);
  pack_weights_kernel<<<8, 256, 0, stream>>>(w2_nl, pw2 + 2048, 64);
  pack_weights_kernel<<<4, 256, 0, stream>>>(w2_pr, pw2 + 4096, 32);
  pack_weights_kernel<<<4, 256, 0, stream>>>(w2_nr, pw2 + 5120, 32);
  // 2) layer-1 scatter-sum with fused degree counting
  {
    long long threads = nE * 16;
    int blocks = (int)((threads + 255) / 256);
    scatter1_kernel<<<blocks, 256, 0, stream>>>(x, posE, posE + nE, aggA, invp, nE);
    scatter1_kernel<<<blocks, 256, 0, stream>>>(x, negE, negE + nE, aggB, invn, nE);
  }
  // 3) counts -> reciprocals (1/max(deg,1))
  recip_kernel<<<(n + 255) / 256, 256, 0, stream>>>(invp, invn, n);
  // 4) dense layer 1 (fp32 WMMA, LDS-staged weights) -> z
  {
    int tiles = (n + 15) / 16;
    int blocks = (tiles + 7) / 8;   // 8 waves (tiles) per 256-thread block
    dense1_kernel<<<blocks, 256, 0, stream>>>(x, aggA, aggB, invp, invn,
        pw1, b1_p, b1_n, z, n);
  }
  // 5) re-zero agg buffers for layer 2
  {
    long long cnt = (long long)n * 128;
    int blocks = (int)((cnt + 255) / 256); if (blocks > 32768) blocks = 32768;
    zero_kernel<<<blocks, 256, 0, stream>>>(aggA, cnt);
  }
  // 6) layer-2 scatter-sum of z halves
  {
    long long threads = nE * 16;
    int blocks = (int)((threads + 255) / 256);
    scatter2_kernel<<<blocks, 256, 0, stream>>>(z, posE, posE + nE, aggA, aggB, 0, nE);
    scatter2_kernel<<<blocks, 256, 0, stream>>>(z, negE, negE + nE, aggA, aggB, 1, nE);
  }
  // 7) dense layer 2 (fp32 WMMA, LDS-staged weights), in-place on d_out
  {
    int tiles = (n + 15) / 16;
    int blocks = (tiles + 7) / 8;
    dense2_kernel<<<blocks, 256, 0, stream>>>(z, aggA, aggB, invp, invn,
        pw2, b2_p, b2_n, n);
  }
}